// ModifiedGAT_73469710565843
// MI455X (gfx1250) — compile-verified
//
#include <hip/hip_runtime.h>
#include <hip/hip_bf16.h>

#define LEAKY 0.2f

typedef float v2f __attribute__((ext_vector_type(2)));
typedef float v8f __attribute__((ext_vector_type(8)));

// ---------------------------------------------------------------------------
// C[M x Nc] = A[M x K] @ W[K x Nc], all row-major fp32, via V_WMMA_F32_16X16X4_F32.
// grid.x = ceil(M/16), grid.y = Nc/64. One wave (32 threads) per block computes a
// 16x64 strip: 4 accumulators share one A fragment per K-step.
// ---------------------------------------------------------------------------
__global__ __launch_bounds__(32)
void gemm_f32_wmma(const float* __restrict__ A, const float* __restrict__ W,
                   float* __restrict__ Cm, int M, int K, int Nc) {
  const int lane = threadIdx.x & 31;
  const int hi   = lane >> 4;     // 0 = lanes 0-15, 1 = lanes 16-31
  const int l    = lane & 15;
  const int m0   = blockIdx.x * 16;
  const int n0   = blockIdx.y * 64;

  int row = m0 + l;
  if (row >= M) row = M - 1;      // clamp; padded rows never stored
  const float* __restrict__ arow = A + (size_t)row * K;

  v8f acc0 = {}, acc1 = {}, acc2 = {}, acc3 = {};
  for (int k = 0; k < K; k += 4) {
    // A fragment (16x4): lanes 0-15 -> K = k,k+1 ; lanes 16-31 -> K = k+2,k+3
    v2f a;
    a.x = arow[k + 2 * hi + 0];
    a.y = arow[k + 2 * hi + 1];
    // B fragments (4x16 each): v0 holds rows k (lanes 0-15) / k+2 (lanes 16-31)
    const float* __restrict__ wr0 = W + (size_t)(k + 2 * hi) * Nc + n0 + l;
    const float* __restrict__ wr1 = wr0 + Nc;
    v2f b0, b1, b2, b3;
    b0.x = wr0[0];  b0.y = wr1[0];
    b1.x = wr0[16]; b1.y = wr1[16];
    b2.x = wr0[32]; b2.y = wr1[32];
    b3.x = wr0[48]; b3.y = wr1[48];
    acc0 = __builtin_amdgcn_wmma_f32_16x16x4_f32(false, a, false, b0, (short)0, acc0, false, false);
    acc1 = __builtin_amdgcn_wmma_f32_16x16x4_f32(false, a, false, b1, (short)0, acc1, false, false);
    acc2 = __builtin_amdgcn_wmma_f32_16x16x4_f32(false, a, false, b2, (short)0, acc2, false, false);
    acc3 = __builtin_amdgcn_wmma_f32_16x16x4_f32(false, a, false, b3, (short)0, acc3, false, false);
  }
  // C/D layout: VGPR r -> (M = r + 8*hi, N = l)
  for (int r = 0; r < 8; ++r) {
    const int rr = m0 + hi * 8 + r;
    if (rr < M) {
      float* __restrict__ co = Cm + (size_t)rr * Nc + n0 + l;
      co[0]  = acc0[r];
      co[16] = acc1[r];
      co[32] = acc2[r];
      co[48] = acc3[r];
    }
  }
}

// ---------------------------------------------------------------------------
__global__ void fill_f32(float* __restrict__ p, float v, long long n) {
  long long i = (long long)blockIdx.x * blockDim.x + threadIdx.x;
  if (i < n) p[i] = v;
}

// als[n*H+h] = dot(h[n, h*C : (h+1)*C], a_s[h]); ald likewise. One thread per (n,h).
__global__ void attn_logits(const float* __restrict__ h, const float* __restrict__ a_s,
                            const float* __restrict__ a_d, float* __restrict__ als,
                            float* __restrict__ ald, int Nn, int Hh, int Cc) {
  int i = blockIdx.x * blockDim.x + threadIdx.x;  // i = n*H + h
  if (i >= Nn * Hh) return;
  int hh = i % Hh;
  const float4* __restrict__ hv  = (const float4*)(h + (size_t)i * Cc);
  const float4* __restrict__ as4 = (const float4*)(a_s + (size_t)hh * Cc);
  const float4* __restrict__ ad4 = (const float4*)(a_d + (size_t)hh * Cc);
  float s = 0.f, d = 0.f;
  for (int c = 0; c < Cc / 4; ++c) {
    float4 x = hv[c], u = as4[c], w = ad4[c];
    s += x.x * u.x + x.y * u.y + x.z * u.z + x.w * u.w;
    d += x.x * w.x + x.y * w.y + x.z * w.z + x.w * w.w;
  }
  als[i] = s;
  ald[i] = d;
}

__device__ __forceinline__ void atomicMaxF(float* addr, float v) {
  // IEEE bit pattern ordering trick: positives order as signed ints,
  // negatives order reversed as unsigned ints.
  if (v >= 0.f)
    atomicMax((int*)addr, __float_as_int(v));
  else
    atomicMin((unsigned int*)addr, (unsigned int)__float_as_int(v));
}

__device__ __forceinline__ void edge_nodes(int e, int Eraw,
                                           const int* __restrict__ srcA,
                                           const int* __restrict__ dstA,
                                           int& s, int& d) {
  if (e < Eraw) { s = srcA[e]; d = dstA[e]; }
  else          { s = e - Eraw; d = s; }         // appended self-loops
}

// segment_max of leaky_relu(al_s[src]+al_d[dst]) into m[dst]. One thread per edge.
__global__ void edge_max(const int* __restrict__ srcA, const int* __restrict__ dstA,
                         const float* __restrict__ als, const float* __restrict__ ald,
                         float* __restrict__ m, int Etot, int Eraw, int Hh) {
  int e = blockIdx.x * blockDim.x + threadIdx.x;
  if (e >= Etot) return;
  int s, d;
  edge_nodes(e, Eraw, srcA, dstA, s, d);
  for (int hh = 0; hh < Hh; ++hh) {
    float v = als[s * Hh + hh] + ald[d * Hh + hh];
    v = v > 0.f ? v : LEAKY * v;
    atomicMaxF(&m[d * Hh + hh], v);
  }
}

// segment_sum of exp(e - m[dst]) into den[dst]. One thread per edge.
__global__ void edge_expsum(const int* __restrict__ srcA, const int* __restrict__ dstA,
                            const float* __restrict__ als, const float* __restrict__ ald,
                            const float* __restrict__ m, float* __restrict__ den,
                            int Etot, int Eraw, int Hh) {
  int e = blockIdx.x * blockDim.x + threadIdx.x;
  if (e >= Etot) return;
  int s, d;
  edge_nodes(e, Eraw, srcA, dstA, s, d);
  for (int hh = 0; hh < Hh; ++hh) {
    float v = als[s * Hh + hh] + ald[d * Hh + hh];
    v = v > 0.f ? v : LEAKY * v;
    atomicAdd(&den[d * Hh + hh], __expf(v - m[d * Hh + hh]));
  }
}

// out[dst] += alpha * h[src] ; one thread handles 4 consecutive channels of one edge.
__global__ void edge_aggr(const int* __restrict__ srcA, const int* __restrict__ dstA,
                          const float* __restrict__ als, const float* __restrict__ ald,
                          const float* __restrict__ m, const float* __restrict__ den,
                          const float* __restrict__ h, float* __restrict__ out,
                          int Etot, int Eraw, int Hh, int Cc) {
  const int HC = Hh * Cc;
  const int Q  = HC >> 2;                       // float4 chunks per edge
  long long idx = (long long)blockIdx.x * blockDim.x + threadIdx.x;
  long long total = (long long)Etot * Q;
  if (idx >= total) return;
  int q = (int)(idx % Q);
  int e = (int)(idx / Q);
  int c0 = q * 4;
  int hh = c0 / Cc;
  int s, d;
  edge_nodes(e, Eraw, srcA, dstA, s, d);
  float v = als[s * Hh + hh] + ald[d * Hh + hh];
  v = v > 0.f ? v : LEAKY * v;
  float alpha = __expf(v - m[d * Hh + hh]) / (den[d * Hh + hh] + 1e-16f);
  const float4 hv = *(const float4*)(h + (size_t)s * HC + c0);
  float* __restrict__ o = out + (size_t)d * HC + c0;
  atomicAdd(o + 0, alpha * hv.x);
  atomicAdd(o + 1, alpha * hv.y);
  atomicAdd(o + 2, alpha * hv.z);
  atomicAdd(o + 3, alpha * hv.w);
}

// out[n,j] = elu(out[n,j] + b[j])
__global__ void bias_elu(float* __restrict__ out, const float* __restrict__ b,
                         long long n, int F) {
  long long i = (long long)blockIdx.x * blockDim.x + threadIdx.x;
  if (i >= n) return;
  float v = out[i] + b[i % F];
  out[i] = v > 0.f ? v : (__expf(v) - 1.f);
}

// ---------------------------------------------------------------------------
static inline dim3 g1(long long n, int b) { return dim3((unsigned)((n + b - 1) / b)); }

static void run_gat_layer(const float* hin, const int* src, const int* dst,
                          const float* W, const float* a_s, const float* a_d,
                          const float* b, float* hbuf, float* outbuf,
                          float* als, float* ald, float* m, float* den,
                          int Nn, int K, int Hh, int Cc, int Eraw, int Etot,
                          hipStream_t stream) {
  const int HC = Hh * Cc;
  // h = hin @ W  ([Nn,K] @ [K,HC])
  dim3 gg((Nn + 15) / 16, HC / 64);
  gemm_f32_wmma<<<gg, 32, 0, stream>>>(hin, W, hbuf, Nn, K, HC);
  // logits
  attn_logits<<<g1((long long)Nn * Hh, 256), 256, 0, stream>>>(hbuf, a_s, a_d, als, ald, Nn, Hh, Cc);
  // init reductions + output accumulator
  fill_f32<<<g1((long long)Nn * Hh, 256), 256, 0, stream>>>(m, -__builtin_huge_valf(), (long long)Nn * Hh);
  fill_f32<<<g1((long long)Nn * Hh, 256), 256, 0, stream>>>(den, 0.f, (long long)Nn * Hh);
  fill_f32<<<g1((long long)Nn * HC, 256), 256, 0, stream>>>(outbuf, 0.f, (long long)Nn * HC);
  // softmax-aggregate over edges
  edge_max<<<g1(Etot, 256), 256, 0, stream>>>(src, dst, als, ald, m, Etot, Eraw, Hh);
  edge_expsum<<<g1(Etot, 256), 256, 0, stream>>>(src, dst, als, ald, m, den, Etot, Eraw, Hh);
  edge_aggr<<<g1((long long)Etot * (HC / 4), 256), 256, 0, stream>>>(
      src, dst, als, ald, m, den, hbuf, outbuf, Etot, Eraw, Hh, Cc);
  // bias + elu
  bias_elu<<<g1((long long)Nn * HC, 256), 256, 0, stream>>>(outbuf, b, (long long)Nn * HC, HC);
}

extern "C" void kernel_launch(void* const* d_in, const int* in_sizes, int n_in,
                              void* d_out, int out_size, void* d_ws, size_t ws_size,
                              hipStream_t stream) {
  (void)n_in; (void)out_size; (void)ws_size;
  const float* x       = (const float*)d_in[0];
  const int*   eidx    = (const int*)d_in[1];
  const float* W0      = (const float*)d_in[2];
  const float* a_src0  = (const float*)d_in[3];
  const float* a_dst0  = (const float*)d_in[4];
  const float* b0      = (const float*)d_in[5];
  const float* W1      = (const float*)d_in[6];
  const float* a_src1  = (const float*)d_in[7];
  const float* a_dst1  = (const float*)d_in[8];
  const float* b1      = (const float*)d_in[9];
  const float* W2      = (const float*)d_in[10];
  const float* a_src2  = (const float*)d_in[11];
  const float* a_dst2  = (const float*)d_in[12];
  const float* b2      = (const float*)d_in[13];

  const int IN   = 256;
  const int Nn   = in_sizes[0] / IN;   // 50000
  const int Eraw = in_sizes[1] / 2;    // 800000
  const int Etot = Eraw + Nn;          // + self-loops
  const int* src = eidx;
  const int* dst = eidx + Eraw;

  // workspace layout (floats)
  float* ws   = (float*)d_ws;
  float* h0   = ws;                         // [N,256]
  float* o0   = h0  + (size_t)Nn * 256;     // [N,256] layer0 out -> layer1 in
  float* h12  = o0  + (size_t)Nn * 256;     // [N,64]  h for layers 1 and 2
  float* o1   = h12 + (size_t)Nn * 64;      // [N,64]  layer1 out -> layer2 in
  float* als  = o1  + (size_t)Nn * 64;      // [N,4]
  float* ald  = als + (size_t)Nn * 4;       // [N,4]
  float* m    = ald + (size_t)Nn * 4;       // [N,4]
  float* den  = m   + (size_t)Nn * 4;       // [N,4]

  // Layer 0: 256 -> 4 heads x 64, concat
  run_gat_layer(x, src, dst, W0, a_src0, a_dst0, b0, h0, o0,
                als, ald, m, den, Nn, 256, 4, 64, Eraw, Etot, stream);
  // Layer 1: 256 -> 64, 1 head (mean over 1 head == identity)
  run_gat_layer(o0, src, dst, W1, a_src1, a_dst1, b1, h12, o1,
                als, ald, m, den, Nn, 256, 1, 64, Eraw, Etot, stream);
  // Layer 2: 64 -> 64, 1 head, result into d_out
  run_gat_layer(o1, src, dst, W2, a_src2, a_dst2, b2, h12, (float*)d_out,
                als, ald, m, den, Nn, 64, 1, 64, Eraw, Etot, stream);
}